// SGATLayer_3186865734207
// MI455X (gfx1250) — compile-verified
//
#include <hip/hip_runtime.h>
#include <hip/hip_bf16.h>

#define N_NODES   100000
#define N_EDGES   1600000
#define IN_DIM    256
#define OUT_DIM   64
#define NEG_SLOPE 0.01f
#define M_TILES   (N_NODES / 16)   // 6250

typedef float v2f __attribute__((ext_vector_type(2)));
typedef float v8f __attribute__((ext_vector_type(8)));

// ---------------------------------------------------------------------------
// 0) init: out = 0, emax = -inf, denom = 0
// ---------------------------------------------------------------------------
__global__ void gat_init(float* __restrict__ out, float* __restrict__ emax,
                         float* __restrict__ denom) {
    int i = blockIdx.x * blockDim.x + threadIdx.x;
    if (i < N_NODES * OUT_DIM) out[i] = 0.0f;
    if (i < N_NODES) { emax[i] = -__builtin_inff(); denom[i] = 0.0f; }
}

// ---------------------------------------------------------------------------
// 1) z = h @ W^T via V_WMMA_F32_16X16X4_F32.
//    Block = 256 threads = 8 waves. Each wave owns one 16-row M-tile and
//    computes all four 16-col N-tiles (A fragment loaded once per k-step).
//    W (64x256 f32 = 64KB) staged in LDS once per block.
// ---------------------------------------------------------------------------
__global__ __launch_bounds__(256)
void gat_gemm_wmma(const float* __restrict__ h, const float* __restrict__ W,
                   float* __restrict__ z) {
    __shared__ float lw[OUT_DIM * IN_DIM];          // 64 KB

    const int tid = threadIdx.x;
    // cooperative load of W into LDS: 16384 floats / 256 threads = 16 float4 each
    {
        const float4* Wv  = (const float4*)W;
        float4*       lwv = (float4*)lw;
#pragma unroll
        for (int i = 0; i < 16; ++i)
            lwv[tid + i * 256] = Wv[tid + i * 256];
    }
    __syncthreads();

    const int wave  = tid >> 5;
    const int lane  = tid & 31;
    const int mTile = blockIdx.x * 8 + wave;
    if (mTile >= M_TILES) return;                   // wave-uniform: EXEC all-1s

    const int   mRow  = mTile * 16 + (lane & 15);   // A row for this lane
    const int   kOff  = (lane < 16) ? 0 : 2;        // K sub-offset in group of 4
    const int   nLane = lane & 15;                  // B/C column for this lane
    const float* __restrict__ hRow = h + (size_t)mRow * IN_DIM;

    v8f acc0 = {}, acc1 = {}, acc2 = {}, acc3 = {};

    for (int k0 = 0; k0 < IN_DIM; k0 += 4) {
        v2f a;
        a.x = hRow[k0 + kOff + 0];
        a.y = hRow[k0 + kOff + 1];

        const float* lwk = lw + k0 + kOff;
        v2f b0, b1, b2, b3;
        b0.x = lwk[(nLane +  0) * IN_DIM];  b0.y = lwk[(nLane +  0) * IN_DIM + 1];
        b1.x = lwk[(nLane + 16) * IN_DIM];  b1.y = lwk[(nLane + 16) * IN_DIM + 1];
        b2.x = lwk[(nLane + 32) * IN_DIM];  b2.y = lwk[(nLane + 32) * IN_DIM + 1];
        b3.x = lwk[(nLane + 48) * IN_DIM];  b3.y = lwk[(nLane + 48) * IN_DIM + 1];

        acc0 = __builtin_amdgcn_wmma_f32_16x16x4_f32(false, a, false, b0, (short)0, acc0, false, false);
        acc1 = __builtin_amdgcn_wmma_f32_16x16x4_f32(false, a, false, b1, (short)0, acc1, false, false);
        acc2 = __builtin_amdgcn_wmma_f32_16x16x4_f32(false, a, false, b2, (short)0, acc2, false, false);
        acc3 = __builtin_amdgcn_wmma_f32_16x16x4_f32(false, a, false, b3, (short)0, acc3, false, false);
    }

    // C/D layout: VGPR r -> row (r + (lane<16?0:8)), col = lane&15
    const int mBase = mTile * 16 + ((lane < 16) ? 0 : 8);
    float* zb = z + (size_t)mBase * OUT_DIM + nLane;
#pragma unroll
    for (int r = 0; r < 8; ++r) {
        zb[(size_t)r * OUT_DIM +  0] = acc0[r];
        zb[(size_t)r * OUT_DIM + 16] = acc1[r];
        zb[(size_t)r * OUT_DIM + 32] = acc2[r];
        zb[(size_t)r * OUT_DIM + 48] = acc3[r];
    }
}

// ---------------------------------------------------------------------------
// 2) s = z @ a_src, d = z @ a_dst. One wave (32 lanes) per node, 2 ch/lane.
// ---------------------------------------------------------------------------
__global__ __launch_bounds__(256)
void gat_sd(const float* __restrict__ z, const float* __restrict__ A,
            float* __restrict__ sbuf, float* __restrict__ dbuf) {
    int gthread = blockIdx.x * blockDim.x + threadIdx.x;
    int node = gthread >> 5;
    int lane = threadIdx.x & 31;
    if (node >= N_NODES) return;

    float2 zv = ((const float2*)(z + (size_t)node * OUT_DIM))[lane];
    float2 as = ((const float2*)A)[lane];              // A[0, 0:64]
    float2 ad = ((const float2*)(A + OUT_DIM))[lane];  // A[0, 64:128]
    float ps = zv.x * as.x + zv.y * as.y;
    float pd = zv.x * ad.x + zv.y * ad.y;
#pragma unroll
    for (int off = 16; off > 0; off >>= 1) {
        ps += __shfl_down(ps, off);
        pd += __shfl_down(pd, off);
    }
    if (lane == 0) { sbuf[node] = ps; dbuf[node] = pd; }
}

// ---------------------------------------------------------------------------
// 3) edge pass A: e = leakyrelu(s[src]+d[dst]); emax[dst] = max(emax, e)
//    signed-magnitude float atomic max (works with -inf init)
// ---------------------------------------------------------------------------
__device__ __forceinline__ void atomicMaxFloat(float* addr, float val) {
    if (val >= 0.0f)
        atomicMax((int*)addr, __float_as_int(val));
    else
        atomicMin((unsigned int*)addr, __float_as_uint(val));
}

__global__ __launch_bounds__(256)
void gat_edge_max(const float* __restrict__ sbuf, const float* __restrict__ dbuf,
                  const int* __restrict__ src, const int* __restrict__ dst,
                  float* __restrict__ e_out, float* __restrict__ emax) {
    int i = blockIdx.x * blockDim.x + threadIdx.x;
    if (i >= N_EDGES) return;
    float e = sbuf[src[i]] + dbuf[dst[i]];
    e = (e >= 0.0f) ? e : NEG_SLOPE * e;
    e_out[i] = e;
    atomicMaxFloat(emax + dst[i], e);
}

// ---------------------------------------------------------------------------
// 4) emax = isfinite(emax) ? emax : 0
// ---------------------------------------------------------------------------
__global__ __launch_bounds__(256)
void gat_fix_emax(float* __restrict__ emax) {
    int i = blockIdx.x * blockDim.x + threadIdx.x;
    if (i >= N_NODES) return;
    float v = emax[i];
    emax[i] = __builtin_isfinite(v) ? v : 0.0f;
}

// ---------------------------------------------------------------------------
// 5) edge pass B: eexp = exp(e - emax[dst]); denom[dst] += eexp
// ---------------------------------------------------------------------------
__global__ __launch_bounds__(256)
void gat_edge_exp(const int* __restrict__ dst, const float* __restrict__ emax,
                  float* __restrict__ e_buf, float* __restrict__ denom) {
    int i = blockIdx.x * blockDim.x + threadIdx.x;
    if (i >= N_EDGES) return;
    int t = dst[i];
    float ex = __expf(e_buf[i] - emax[t]);
    e_buf[i] = ex;
    atomicAdd(denom + t, ex);
}

// ---------------------------------------------------------------------------
// 6) scatter: out[dst] += (eexp/denom[dst]) * z[src]. 32 lanes/edge, 2 ch/lane.
// ---------------------------------------------------------------------------
__global__ __launch_bounds__(256)
void gat_scatter(const float* __restrict__ z, const int* __restrict__ src,
                 const int* __restrict__ dst, const float* __restrict__ e_buf,
                 const float* __restrict__ denom, float* __restrict__ out) {
    int gthread = blockIdx.x * blockDim.x + threadIdx.x;
    int edge = gthread >> 5;
    int lane = threadIdx.x & 31;
    if (edge >= N_EDGES) return;
    int sN = src[edge], tN = dst[edge];
    float den = denom[tN];
    float w = e_buf[edge] / ((den > 0.0f) ? den : 1.0f);
    float2 zv = ((const float2*)(z + (size_t)sN * OUT_DIM))[lane];
    float* ob = out + (size_t)tN * OUT_DIM + 2 * lane;
    atomicAdd(ob + 0, w * zv.x);
    atomicAdd(ob + 1, w * zv.y);
}

// ---------------------------------------------------------------------------
extern "C" void kernel_launch(void* const* d_in, const int* in_sizes, int n_in,
                              void* d_out, int out_size, void* d_ws, size_t ws_size,
                              hipStream_t stream) {
    const float* h   = (const float*)d_in[0];
    const float* W   = (const float*)d_in[1];
    const float* A   = (const float*)d_in[2];
    const int*   src = (const int*)d_in[3];
    const int*   dst = (const int*)d_in[4];
    float* out = (float*)d_out;

    float* ws    = (float*)d_ws;
    float* z     = ws;                 // 6,400,000 floats
    float* sbuf  = ws + 6400000;       //   100,000
    float* dbuf  = ws + 6500000;       //   100,000
    float* emax  = ws + 6600000;       //   100,000
    float* denom = ws + 6700000;       //   100,000
    float* eexp  = ws + 6800000;       // 1,600,000   (total ~33.6 MB)

    gat_init<<<(N_NODES * OUT_DIM + 255) / 256, 256, 0, stream>>>(out, emax, denom);
    gat_gemm_wmma<<<(M_TILES + 7) / 8, 256, 0, stream>>>(h, W, z);
    gat_sd<<<(N_NODES * 32 + 255) / 256, 256, 0, stream>>>(z, A, sbuf, dbuf);
    gat_edge_max<<<(N_EDGES + 255) / 256, 256, 0, stream>>>(sbuf, dbuf, src, dst, eexp, emax);
    gat_fix_emax<<<(N_NODES + 255) / 256, 256, 0, stream>>>(emax);
    gat_edge_exp<<<(N_EDGES + 255) / 256, 256, 0, stream>>>(dst, emax, eexp, denom);
    gat_scatter<<<(N_EDGES * 32 + 255) / 256, 256, 0, stream>>>(z, src, dst, eexp, denom, out);
}